// NSVQ_78176994722627
// MI455X (gfx1250) — compile-verified
//
#include <hip/hip_runtime.h>
#include <hip/hip_bf16.h>

typedef float v2f __attribute__((ext_vector_type(2)));
typedef float v8f __attribute__((ext_vector_type(8)));

#define D_DIM 64
#define KSTEPS 16        // 16 WMMA steps of K=4 -> K=64
#define STAGE_CODES 64   // codes staged into LDS per double-buffer step
#define LDS_STRIDE 66    // row stride (floats) to kill 16-way bank conflicts

// async DMA: copy 16B per lane, global -> LDS; IOFFSET applies to both addresses
#define ASYNC_CP16(ldsb, gaddr, OFF)                                          \
    asm volatile("global_load_async_to_lds_b128 %0, %1, off offset:" #OFF     \
                 :: "v"(ldsb), "v"(gaddr) : "memory")

#define WAIT_ASYNC0() asm volatile("s_wait_asynccnt 0x0" ::: "memory")

// ---------------------------------------------------------------------------
// Kernel A: precompute ||W[j]||^2 for all codes
// ---------------------------------------------------------------------------
__global__ __launch_bounds__(256) void nsvq_wnorm_kernel(
    const float* __restrict__ W, float* __restrict__ ww, int n_e)
{
    const int j = blockIdx.x * 256 + threadIdx.x;
    if (j < n_e) {
        const float4* r = (const float4*)(W + (size_t)j * D_DIM);
        float s = 0.0f;
#pragma unroll
        for (int i = 0; i < 16; ++i) {
            const float4 v = r[i];
            s += v.x * v.x + v.y * v.y + v.z * v.z + v.w * v.w;
        }
        ww[j] = s;
    }
}

// ---------------------------------------------------------------------------
// Kernel 1: argmin via WMMA f32 16x16x4, W staged in LDS by async-to-LDS DMA.
// grid: N/128 blocks of 256 threads (8 waves); each wave owns 16 rows of z.
// All 8 waves share each staged 64-code tile (8x global-traffic reduction).
// ---------------------------------------------------------------------------
__global__ __launch_bounds__(256) void nsvq_argmin_kernel(
    const float* __restrict__ z, const float* __restrict__ W,
    const float* __restrict__ ww, int n_e,
    int* __restrict__ idx_i, float* __restrict__ idx_f)
{
    __shared__ float Wlds[2][STAGE_CODES * LDS_STRIDE];
    __shared__ float wwlds[2][STAGE_CODES];

    const int tid  = threadIdx.x;
    const int lane = tid & 31;
    const int wave = tid >> 5;
    const int m    = lane & 15;   // M (A) / N (B,D) position of this lane
    const int half = lane >> 4;   // k-pair select (A/B), M+8 select (D)
    const int row0 = blockIdx.x * 128 + wave * 16;

    // Preload A tile: row (row0+m), k-pairs {4t+2*half, 4t+2*half+1}
    const float* zr = z + (size_t)(row0 + m) * D_DIM + half * 2;
    v2f a[KSTEPS];
#pragma unroll
    for (int t = 0; t < KSTEPS; ++t) {
        const float2 v = *(const float2*)(zr + t * 4);
        a[t].x = v.x; a[t].y = v.y;
    }

    // per-thread DMA assignment: row r of the stage, 16-float quarter cq
    const int r  = tid >> 2;
    const int cq = (tid & 3) * 16;
    const uint32_t ldsW0 = (uint32_t)(uintptr_t)&Wlds[0][r * LDS_STRIDE + cq];
    const uint32_t ldsW1 = (uint32_t)(uintptr_t)&Wlds[1][r * LDS_STRIDE + cq];
    const uint32_t ldsN0 = (uint32_t)(uintptr_t)&wwlds[0][(tid & 15) * 4];
    const uint32_t ldsN1 = (uint32_t)(uintptr_t)&wwlds[1][(tid & 15) * 4];

    float minv[8];
    int   mini[8];
#pragma unroll
    for (int v = 0; v < 8; ++v) { minv[v] = 3.4e38f; mini[v] = 0; }

    const int nstage = n_e / STAGE_CODES;

    // prologue: DMA stage 0 into buffer 0
    {
        const unsigned long long g =
            (unsigned long long)(W + (size_t)r * D_DIM + cq);
        ASYNC_CP16(ldsW0, g, 0);  ASYNC_CP16(ldsW0, g, 16);
        ASYNC_CP16(ldsW0, g, 32); ASYNC_CP16(ldsW0, g, 48);
        if (tid < 16) {
            const unsigned long long gn =
                (unsigned long long)(ww + (size_t)tid * 4);
            ASYNC_CP16(ldsN0, gn, 0);
        }
    }

    for (int s = 0; s < nstage; ++s) {
        const int cur = s & 1;
        const int c0  = s * STAGE_CODES;

        WAIT_ASYNC0();      // my async copies for buf[cur] have landed
        __syncthreads();    // everyone's copies landed; buf[cur^1] is free

        // prefetch next stage into the other buffer (overlaps with WMMA below)
        if (s + 1 < nstage) {
            const int nc0 = c0 + STAGE_CODES;
            const unsigned long long g =
                (unsigned long long)(W + (size_t)(nc0 + r) * D_DIM + cq);
            const uint32_t lw = cur ? ldsW0 : ldsW1;
            ASYNC_CP16(lw, g, 0);  ASYNC_CP16(lw, g, 16);
            ASYNC_CP16(lw, g, 32); ASYNC_CP16(lw, g, 48);
            if (tid < 16) {
                const unsigned long long gn =
                    (unsigned long long)(ww + (size_t)(nc0 + tid * 4));
                ASYNC_CP16(cur ? ldsN0 : ldsN1, gn, 0);
            }
        }

        // compute: 4 col-tiles of 16 codes, B fragments streamed from LDS.
        // Load the full B tile first so all ds_loads are in flight before the
        // WMMA chain starts (DS returns in-order -> staggered dscnt waits).
#pragma unroll
        for (int ct = 0; ct < 4; ++ct) {
            const float* Bl = &Wlds[cur][(ct * 16 + m) * LDS_STRIDE + half * 2];
            const float wwv = wwlds[cur][ct * 16 + m];
            v2f b[KSTEPS];
#pragma unroll
            for (int t = 0; t < KSTEPS; ++t) {
                const float2 bv = *(const float2*)(Bl + t * 4);  // ds_load_b64
                b[t].x = bv.x; b[t].y = bv.y;
            }
            v8f c = {};
#pragma unroll
            for (int t = 0; t < KSTEPS; ++t) {
                c = __builtin_amdgcn_wmma_f32_16x16x4_f32(
                        false, a[t], false, b[t], (short)0, c, false, false);
            }
            // score = ||w||^2 - 2*dot  (||z||^2 is row-constant -> dropped)
            const int n = c0 + ct * 16 + m;
#pragma unroll
            for (int v = 0; v < 8; ++v) {
                const float sc = fmaf(-2.0f, c[v], wwv);
                if (sc < minv[v]) { minv[v] = sc; mini[v] = n; }
            }
        }
        __syncthreads();    // all waves done with buf[cur] before it is refilled
    }

    // Butterfly min-reduce across the 16 N-lanes of each half
#pragma unroll
    for (int off = 8; off >= 1; off >>= 1) {
#pragma unroll
        for (int v = 0; v < 8; ++v) {
            const float ov = __shfl_xor(minv[v], off, 32);
            const int   oi = __shfl_xor(mini[v], off, 32);
            if (ov < minv[v] || (ov == minv[v] && oi < mini[v])) {
                minv[v] = ov; mini[v] = oi;
            }
        }
    }

    if (m == 0) {  // lane 0 -> rows row0..+7, lane 16 -> rows row0+8..+15
#pragma unroll
        for (int v = 0; v < 8; ++v) {
            const int row = row0 + half * 8 + v;
            idx_i[row] = mini[v];
            idx_f[row] = (float)mini[v];
        }
    }
}

// ---------------------------------------------------------------------------
// Kernel 2: gather z_q = W[idx], accumulate SSE and usage counts
// ---------------------------------------------------------------------------
__global__ __launch_bounds__(256) void nsvq_gather_kernel(
    const float* __restrict__ z, const float* __restrict__ W,
    const int* __restrict__ idx_i, float* __restrict__ zq_out,
    float* __restrict__ counts, float* __restrict__ sse)
{
    __shared__ float red[256];
    const int e   = blockIdx.x * 256 + threadIdx.x;
    const int row = e >> 6;
    const int d   = e & 63;
    const int k   = idx_i[row];
    const float q = W[(size_t)k * D_DIM + d];
    zq_out[e] = q;                       // STE forward value == z_q
    if (d == 0) atomicAdd(&counts[k], 1.0f);
    const float diff = q - z[e];
    red[threadIdx.x] = diff * diff;
    __syncthreads();
    for (int s = 128; s > 0; s >>= 1) {
        if (threadIdx.x < s) red[threadIdx.x] += red[threadIdx.x + s];
        __syncthreads();
    }
    if (threadIdx.x == 0) atomicAdd(sse, red[0]);
}

// ---------------------------------------------------------------------------
// Kernel 0: zero the accumulators (deterministic across graph replays)
// ---------------------------------------------------------------------------
__global__ void nsvq_zero_kernel(float* __restrict__ p, int n)
{
    const int i = blockIdx.x * blockDim.x + threadIdx.x;
    if (i < n) p[i] = 0.0f;
}

// ---------------------------------------------------------------------------
// Kernel 3: loss = 1.25 * SSE / (N*D); perplexity from usage histogram
// ---------------------------------------------------------------------------
__global__ __launch_bounds__(256) void nsvq_finalize_kernel(
    const float* __restrict__ counts, const float* __restrict__ sse,
    int n_e, float invN, float invND,
    float* __restrict__ loss_out, float* __restrict__ perp_out)
{
    __shared__ float red[256];
    float s = 0.0f;
    for (int i = threadIdx.x; i < n_e; i += 256) {
        const float p = counts[i] * invN;
        s += p * logf(p + 1e-12f);
    }
    red[threadIdx.x] = s;
    __syncthreads();
    for (int st = 128; st > 0; st >>= 1) {
        if (threadIdx.x < st) red[threadIdx.x] += red[threadIdx.x + st];
        __syncthreads();
    }
    if (threadIdx.x == 0) {
        *perp_out = expf(-red[0]);
        *loss_out = 1.25f * (*sse) * invND;   // (1 + BETA) * MSE
    }
}

extern "C" void kernel_launch(void* const* d_in, const int* in_sizes, int n_in,
                              void* d_out, int out_size, void* d_ws, size_t ws_size,
                              hipStream_t stream)
{
    const float* z = (const float*)d_in[0];   // [B*T*D] f32
    const float* W = (const float*)d_in[1];   // [n_e*D] f32
    const int nz  = in_sizes[0];              // B*T*D = 2097152
    const int nw  = in_sizes[1];              // n_e*D = 524288
    const int N   = nz / D_DIM;               // 32768 rows
    const int n_e = nw / D_DIM;               // 8192 codes

    // d_out layout: [z_q (nz)] [loss (1)] [idx as f32 (N)] [perplexity (1)]
    float* out      = (float*)d_out;
    float* zq_out   = out;
    float* loss_out = out + nz;
    float* idx_f    = out + nz + 1;
    float* perp_out = out + nz + 1 + N;

    // workspace: [idx_i: N ints] [counts: n_e f32] [sse: 1 f32] [ww: n_e f32]
    int*   idx_i  = (int*)d_ws;
    float* counts = (float*)((char*)d_ws + (size_t)N * sizeof(int));
    float* sse    = counts + n_e;
    float* ww     = sse + 1;

    nsvq_zero_kernel<<<(n_e + 1 + 255) / 256, 256, 0, stream>>>(counts, n_e + 1);
    nsvq_wnorm_kernel<<<(n_e + 255) / 256, 256, 0, stream>>>(W, ww, n_e);
    nsvq_argmin_kernel<<<N / 128, 256, 0, stream>>>(z, W, ww, n_e, idx_i, idx_f);
    nsvq_gather_kernel<<<(N * D_DIM) / 256, 256, 0, stream>>>(z, W, idx_i, zq_out, counts, sse);
    nsvq_finalize_kernel<<<1, 256, 0, stream>>>(counts, sse, n_e,
                                                1.0f / (float)N, 1.0f / (float)nz,
                                                loss_out, perp_out);
}